// QuartetLoss_77403900608664
// MI455X (gfx1250) — compile-verified
//
#include <hip/hip_runtime.h>
#include <math.h>

typedef __attribute__((ext_vector_type(2))) float v2f;
typedef __attribute__((ext_vector_type(8))) float v8f;

#define EPS 1e-8f

// ---------------------------------------------------------------------------
// Kernel 1: row-normalize features. One wave32 per row of 256 floats
// (8 floats/lane as two float4 loads), shuffle-tree reduce, scale, store.
// ---------------------------------------------------------------------------
__global__ void qk_normalize(const float* __restrict__ x, float* __restrict__ fn,
                             int bs) {
    const int gw   = (blockIdx.x * blockDim.x + threadIdx.x) >> 5;  // wave id == row
    const int lane = threadIdx.x & 31;
    if (gw >= bs) return;

    const float4* row = (const float4*)(x + (size_t)gw * 256);
    float4 a = row[lane * 2 + 0];
    float4 b = row[lane * 2 + 1];
    float ss = a.x * a.x + a.y * a.y + a.z * a.z + a.w * a.w +
               b.x * b.x + b.y * b.y + b.z * b.z + b.w * b.w;
#pragma unroll
    for (int off = 16; off > 0; off >>= 1) ss += __shfl_xor(ss, off, 32);

    const float inv = 1.0f / fmaxf(sqrtf(ss), EPS);
    a.x *= inv; a.y *= inv; a.z *= inv; a.w *= inv;
    b.x *= inv; b.y *= inv; b.z *= inv; b.w *= inv;

    float4* o = (float4*)(fn + (size_t)gw * 256);
    o[lane * 2 + 0] = a;
    o[lane * 2 + 1] = b;
}

// ---------------------------------------------------------------------------
// Kernel 2: pos[n] = dot(fn[n], fn[n+N]); one wave32 per n.
// ---------------------------------------------------------------------------
__global__ void qk_pos(const float* __restrict__ fn, float* __restrict__ pos,
                       int N) {
    const int gw   = (blockIdx.x * blockDim.x + threadIdx.x) >> 5;
    const int lane = threadIdx.x & 31;
    if (gw >= N) return;

    const float4* ra = (const float4*)(fn + (size_t)gw * 256);
    const float4* rb = (const float4*)(fn + (size_t)(gw + N) * 256);
    float s = 0.0f;
#pragma unroll
    for (int j = 0; j < 2; ++j) {
        float4 a = ra[lane * 2 + j];
        float4 b = rb[lane * 2 + j];
        s += a.x * b.x + a.y * b.y + a.z * b.z + a.w * b.w;
    }
#pragma unroll
    for (int off = 16; off > 0; off >>= 1) s += __shfl_xor(s, off, 32);
    if (lane == 0) pos[gw] = s;
}

// ---------------------------------------------------------------------------
// Kernel 3: per-n max over 40 pair cosine sims via V_WMMA_F32_16X16X4_F32.
// Block = 3 waves; wave g handles pairs [g*16, g*16+16) (clamped to m-1; the
// clamp duplicates pair 39, which cannot change the max). A-rows = fn[a_i],
// B-cols = fn[b_i]; K walks D=256 in steps of 4; diagonal of D = sims.
//
// Per CDNA5 ISA layouts (32-bit, 16x4 A / 4x16 B): lane&15 selects the
// matrix row/col, lanes>=16 carry K+2..K+3, so every lane streams float2 at
// rowbase + k + ((lane>>4)<<1). Diagonal of the f32 16x16 D: lanes 0..7 hold
// D[l][l] in acc[l]; lanes 24..31 hold D[l-16][l-16] in acc[l-24].
// ---------------------------------------------------------------------------
__global__ void qk_simsmax(const float* __restrict__ fn,
                           const long long* __restrict__ pair_idx,
                           float* __restrict__ best, int m) {
    const int n    = blockIdx.x;
    const int g    = threadIdx.x >> 5;   // wave in block: 0..2
    const int lane = threadIdx.x & 31;
    const int di   = lane & 15;          // matrix row (A) / col (B) this lane owns

    int p = g * 16 + di;
    if (p > m - 1) p = m - 1;            // clamp -> duplicates, max-safe

    const long long* pp = pair_idx + ((size_t)n * m + p) * 2;
    const int rowA = (int)pp[0];
    const int rowB = (int)pp[1];

    const int koff = (lane >> 4) << 1;   // lanes 16..31 hold K+2,K+3
    const float* A = fn + (size_t)rowA * 256 + koff;
    const float* B = fn + (size_t)rowB * 256 + koff;

    v8f acc = {};
#pragma unroll 8
    for (int k = 0; k < 256; k += 4) {
        v2f a = *(const v2f*)(A + k);
        v2f b = *(const v2f*)(B + k);
        acc = __builtin_amdgcn_wmma_f32_16x16x4_f32(
            /*neg_a=*/false, a, /*neg_b=*/false, b,
            /*c_mod=*/(short)0, acc, /*reuse_a=*/false, /*reuse_b=*/false);
    }

    // pull the diagonal of D out of the accumulator
    float v = -INFINITY;
    if (lane < 8)        v = acc[lane];
    else if (lane >= 24) v = acc[lane - 24];

#pragma unroll
    for (int off = 16; off > 0; off >>= 1) v = fmaxf(v, __shfl_xor(v, off, 32));

    __shared__ float red[3];
    if (lane == 0) red[g] = v;
    __syncthreads();
    if (threadIdx.x == 0)
        best[n] = fmaxf(fmaxf(red[0], red[1]), red[2]);
}

// ---------------------------------------------------------------------------
// Kernel 4: loss = mean(sigmoid(best - pos)); single 256-thread block.
// ---------------------------------------------------------------------------
__global__ void qk_loss(const float* __restrict__ best,
                        const float* __restrict__ pos,
                        float* __restrict__ out, int N) {
    __shared__ float s[256];
    float sum = 0.0f;
    for (int i = threadIdx.x; i < N; i += 256) {
        const float d = best[i] - pos[i];
        sum += 1.0f / (1.0f + __expf(-d));
    }
    s[threadIdx.x] = sum;
    __syncthreads();
#pragma unroll
    for (int off = 128; off > 0; off >>= 1) {
        if (threadIdx.x < off) s[threadIdx.x] += s[threadIdx.x + off];
        __syncthreads();
    }
    if (threadIdx.x == 0) out[0] = s[0] / (float)N;
}

// ---------------------------------------------------------------------------
extern "C" void kernel_launch(void* const* d_in, const int* in_sizes, int n_in,
                              void* d_out, int out_size, void* d_ws, size_t ws_size,
                              hipStream_t stream) {
    const float*     features = (const float*)d_in[0];
    const long long* pair_idx = (const long long*)d_in[1];   // int64 indices

    const int D  = 256;                       // featureD (fixed by setup)
    const int m  = 40;                        // pairs per anchor (fixed by setup)
    const int bs = in_sizes[0] / D;           // 8192
    const int N  = in_sizes[1] / (2 * m);     // 4096

    float* fn   = (float*)d_ws;               // bs*D floats (8 MB)
    float* best = fn + (size_t)bs * D;        // N floats
    float* pos  = best + N;                   // N floats
    float* out  = (float*)d_out;

    // 1) normalize: one wave per row, 8 waves per 256-thread block
    qk_normalize<<<(bs + 7) / 8, 256, 0, stream>>>(features, fn, bs);
    // 2) anchor-pair dots
    qk_pos<<<(N + 7) / 8, 256, 0, stream>>>(fn, pos, N);
    // 3) WMMA pair sims + per-n max (3 waves per block, one block per n)
    qk_simsmax<<<N, 96, 0, stream>>>(fn, pair_idx, best, m);
    // 4) mean(sigmoid(best - pos))
    qk_loss<<<1, 256, 0, stream>>>(best, pos, out, N);
}